// AutoregressiveLayer_5806795784821
// MI455X (gfx1250) — compile-verified
//
#include <hip/hip_runtime.h>

typedef __attribute__((ext_vector_type(2))) float v2f;
typedef __attribute__((ext_vector_type(8))) float v8f;

constexpr int kD  = 64;     // in_features
constexpr int kU  = 1024;   // hidden
constexpr int kB  = 4096;   // batch
constexpr int kBT = 32;     // batch rows per workgroup (2 WMMA M-tiles)

constexpr int kXStride = kD + 4;    // 68, LDS pad to dodge bank conflicts
constexpr int kHStride = kU + 4;    // 1028

// workspace layout (floats)
constexpr int kWaOff = 0;                 // swizzled masked W0  [64 Ntile][8 kpair][32 lane][4]
constexpr int kWaSz  = kU * kD;           // 65536
constexpr int kWbOff = kWaOff + kWaSz;    // swizzled masked W1  [64][128][32][4]
constexpr int kWbSz  = kU * kU;           // 1048576
constexpr int kWcOff = kWbOff + kWbSz;    // masked W2, plain row-major [128][1024]
constexpr int kWcSz  = 2 * kD * kU;       // 131072

// ---- weight preparation: apply MADE masks + repack into WMMA-ready layout ----
// B-fragment convention for V_WMMA_F32_16X16X4_F32 (B is K=4 x N=16):
//   lane l: VGPR0 = B[(l>>4)*2 + 0][l&15], VGPR1 = B[(l>>4)*2 + 1][l&15]
// One float4 per lane packs a K-chunk pair (chunks 2cp, 2cp+1):
//   q=0,1 -> chunk 2cp k = 8cp + (l>>4)*2 + q ; q=2,3 -> chunk 2cp+1 k = 8cp+4 + (l>>4)*2 + (q-2)

__global__ void prep_w0(const float* __restrict__ W0, float* __restrict__ Wa) {
  int e = blockIdx.x * 256 + threadIdx.x;          // e < U*D
  int j = e >> 6, d = e & 63;
  int dh = j % (kD - 1);
  float v = (d <= dh) ? W0[e] : 0.f;
  int t = j >> 4, n = j & 15;
  int cp = d >> 3, rr = d & 7;
  int q  = ((rr >> 2) << 1) | (rr & 1);
  int hi = (rr >> 1) & 1;
  int l  = (hi << 4) | n;
  Wa[(((t * 8 + cp) * 32 + l) << 2) + q] = v;
}

__global__ void prep_w1(const float* __restrict__ W1, float* __restrict__ Wb) {
  int e = blockIdx.x * 256 + threadIdx.x;          // e < U*U
  int j = e >> 10, k = e & 1023;
  int dhj = j % (kD - 1), dhk = k % (kD - 1);
  float v = (dhk <= dhj) ? W1[e] : 0.f;
  int t = j >> 4, n = j & 15;
  int cp = k >> 3, rr = k & 7;
  int q  = ((rr >> 2) << 1) | (rr & 1);
  int hi = (rr >> 1) & 1;
  int l  = (hi << 4) | n;
  Wb[(((t * 128 + cp) * 32 + l) << 2) + q] = v;
}

__global__ void prep_w2(const float* __restrict__ W2, float* __restrict__ Wc) {
  int e = blockIdx.x * 256 + threadIdx.x;          // e < 2*D*U
  int r = e >> 10, k = e & 1023;
  int dhk = k % (kD - 1);
  Wc[e] = (dhk < (r & (kD - 1))) ? W2[e] : 0.f;
}

// ---- main recurrence kernel: one workgroup owns 32 batch rows for all 64 steps ----

__device__ __forceinline__ v8f wmma_f32(v2f a, v2f b, v8f c) {
  return __builtin_amdgcn_wmma_f32_16x16x4_f32(false, a, false, b, (short)0, c,
                                               false, false);
}

__launch_bounds__(256, 1)
__global__ void made_flow_kernel(const float* __restrict__ u,
                                 const float* __restrict__ b0,
                                 const float* __restrict__ b1,
                                 const float* __restrict__ b2,
                                 const float* __restrict__ ws,
                                 float* __restrict__ out) {
  extern __shared__ float smem[];
  float* xs   = smem;                         // [32][68]   current x tile
  float* h0s  = xs  + kBT * kXStride;         // [32][1028] hidden 1
  float* h1s  = h0s + kBT * kHStride;         // [32][1028] hidden 2
  float* red  = h1s + kBT * kHStride;         // [32][8][2] layer-3 partials
  float* ldjs = red + kBT * 8 * 2;            // [32]

  const int tid  = threadIdx.x;
  const int wg   = blockIdx.x;
  const int lane = tid & 31;
  const int wv   = tid >> 5;                  // 8 waves
  const int n    = lane & 15;
  const int hi2  = (lane >> 4) << 1;
  const int hi8  = (lane >> 4) << 3;

  const float* Wa = ws + kWaOff;
  const float* Wb = ws + kWbOff;
  const float* Wc = ws + kWcOff;

  for (int e = tid; e < kBT * kXStride; e += 256) xs[e] = 0.f;
  if (tid < kBT) ldjs[tid] = 0.f;
  __syncthreads();

  for (int i = 0; i < kD; ++i) {
    // ---------- layer 1: h0 = relu(x @ mW0^T + b0), M=32 K=64 N=1024 ----------
    for (int tt = 0; tt < 8; ++tt) {
      const int t = wv * 8 + tt;              // N-tile (16 hidden units)
      const float bias = b0[t * 16 + n];
      v8f acc0, acc1;
      for (int v = 0; v < 8; ++v) { acc0[v] = bias; acc1[v] = bias; }
      const float4* wq = ((const float4*)Wa) + (t * 8) * 32 + lane;
      for (int cp = 0; cp < 8; ++cp) {
        float4 bq = wq[cp * 32];              // global_load_b128, coalesced
        v2f bA = {bq.x, bq.y};
        v2f bB = {bq.z, bq.w};
        const int kb = cp * 8 + hi2;
        v2f a00 = *(const v2f*)&xs[n * kXStride + kb];
        v2f a01 = *(const v2f*)&xs[n * kXStride + kb + 4];
        v2f a10 = *(const v2f*)&xs[(n + 16) * kXStride + kb];
        v2f a11 = *(const v2f*)&xs[(n + 16) * kXStride + kb + 4];
        acc0 = wmma_f32(a00, bA, acc0);
        acc1 = wmma_f32(a10, bA, acc1);       // reuse B fragment for 2nd M-tile
        acc0 = wmma_f32(a01, bB, acc0);
        acc1 = wmma_f32(a11, bB, acc1);
      }
      for (int v = 0; v < 8; ++v) {
        float f0 = acc0[v] > 0.f ? acc0[v] : 0.f;
        float f1 = acc1[v] > 0.f ? acc1[v] : 0.f;
        h0s[(v + hi8) * kHStride + t * 16 + n]        = f0;
        h0s[(16 + v + hi8) * kHStride + t * 16 + n]   = f1;
      }
    }
    __syncthreads();

    // ---------- layer 2: h1 = relu(h0 @ mW1^T + b1), M=32 K=1024 N=1024 ----------
    for (int tt = 0; tt < 8; ++tt) {
      const int t = wv * 8 + tt;
      const float bias = b1[t * 16 + n];
      v8f acc0, acc1;
      for (int v = 0; v < 8; ++v) { acc0[v] = bias; acc1[v] = bias; }
      const float4* wq = ((const float4*)Wb) + (t * 128) * 32 + lane;
      for (int cp = 0; cp < 128; ++cp) {
        float4 bq = wq[cp * 32];
        v2f bA = {bq.x, bq.y};
        v2f bB = {bq.z, bq.w};
        const int kb = cp * 8 + hi2;
        v2f a00 = *(const v2f*)&h0s[n * kHStride + kb];
        v2f a01 = *(const v2f*)&h0s[n * kHStride + kb + 4];
        v2f a10 = *(const v2f*)&h0s[(n + 16) * kHStride + kb];
        v2f a11 = *(const v2f*)&h0s[(n + 16) * kHStride + kb + 4];
        acc0 = wmma_f32(a00, bA, acc0);
        acc1 = wmma_f32(a10, bA, acc1);
        acc0 = wmma_f32(a01, bB, acc0);
        acc1 = wmma_f32(a11, bB, acc1);
      }
      for (int v = 0; v < 8; ++v) {
        float f0 = acc0[v] > 0.f ? acc0[v] : 0.f;
        float f1 = acc1[v] > 0.f ? acc1[v] : 0.f;
        h1s[(v + hi8) * kHStride + t * 16 + n]        = f0;
        h1s[(16 + v + hi8) * kHStride + t * 16 + n]   = f1;
      }
    }
    __syncthreads();

    // ---------- layer 3: only rows i (mu) and D+i (sigma) of z ----------
    {
      const int r = tid & 31;                 // batch row in tile
      const int s = tid >> 5;                 // k segment, 128 wide
      const float* wmu = Wc + i * kU;
      const float* wsg = Wc + (kD + i) * kU;
      const float* hrow = h1s + r * kHStride;
      float mp = 0.f, sp = 0.f;
      const int k0 = s * 128;
      for (int k = k0; k < k0 + 128; ++k) {
        float h = hrow[k];
        mp = fmaf(h, wmu[k], mp);
        sp = fmaf(h, wsg[k], sp);
      }
      red[(r * 8 + s) * 2 + 0] = mp;
      red[(r * 8 + s) * 2 + 1] = sp;
    }
    __syncthreads();
    if (tid < kBT) {
      float mu = b2[i], sg = b2[kD + i];
      for (int s = 0; s < 8; ++s) {
        mu += red[(tid * 8 + s) * 2 + 0];
        sg += red[(tid * 8 + s) * 2 + 1];
      }
      const float uu = u[(wg * kBT + tid) * kD + i];
      xs[tid * kXStride + i] = uu * expf(sg) + mu;
      ldjs[tid] += sg;
    }
    __syncthreads();
  }

  // ---------- writeout: x then log_det_jacobian ----------
  for (int e = tid; e < kBT * kD; e += 256) {
    int r = e >> 6, d = e & 63;
    out[(wg * kBT + r) * kD + d] = xs[r * kXStride + d];
  }
  if (tid < kBT) out[kB * kD + wg * kBT + tid] = ldjs[tid];
}

extern "C" void kernel_launch(void* const* d_in, const int* in_sizes, int n_in,
                              void* d_out, int out_size, void* d_ws, size_t ws_size,
                              hipStream_t stream) {
  const float* u  = (const float*)d_in[0];
  const float* W0 = (const float*)d_in[1];
  const float* b0 = (const float*)d_in[2];
  const float* W1 = (const float*)d_in[3];
  const float* b1 = (const float*)d_in[4];
  const float* W2 = (const float*)d_in[5];
  const float* b2 = (const float*)d_in[6];
  float* ws  = (float*)d_ws;
  float* out = (float*)d_out;

  prep_w0<<<(kU * kD) / 256,     256, 0, stream>>>(W0, ws + kWaOff);
  prep_w1<<<(kU * kU) / 256,     256, 0, stream>>>(W1, ws + kWbOff);
  prep_w2<<<(2 * kD * kU) / 256, 256, 0, stream>>>(W2, ws + kWcOff);

  size_t smem = (size_t)(kBT * kXStride + 2 * kBT * kHStride + kBT * 8 * 2 + kBT) *
                sizeof(float);   // ~274 KB dynamic LDS (<= 320 KB/WGP)
  made_flow_kernel<<<kB / kBT, 256, smem, stream>>>(u, b0, b1, b2, ws, out);
}